// UniformMatcher_77841987272886
// MI455X (gfx1250) — compile-verified
//
#include <hip/hip_runtime.h>
#include <stdint.h>

// Problem constants (match the reference harness)
#define BSZ     16
#define QN      32768
#define GN      64
#define KK      4
#define CHUNKS  32
#define CQ      (QN / CHUNKS)   // 1024 queries per block
#define SQ      256             // queries per LDS stage
#define NST     (CQ / SQ)       // 4 stages
#define NTASK   (BSZ * 2 * GN)  // 2048 (b, type, g) reduction tasks

// ---- CDNA5 async global->LDS path (ASYNCcnt-tracked) ----------------------

// Low 32 bits of a generic pointer into LDS == wave-relative LDS byte offset
// (ISA 10.2: only the low 32 address bits are used).
__device__ __forceinline__ unsigned lds_off32(const void* p) {
  return (unsigned)(unsigned long long)p;
}

// GLOBAL_LOAD_ASYNC_TO_LDS_B128: per-lane 16B copy, GV addressing (64-bit vaddr).
__device__ __forceinline__ void async_copy_b128(unsigned lds_addr, const float* gptr) {
  asm volatile("global_load_async_to_lds_b128 %0, %1, off"
               :: "v"(lds_addr), "v"((unsigned long long)gptr)
               : "memory");
}

__device__ __forceinline__ void wait_async_le(int) {} // doc helper (unused)

__device__ __forceinline__ void s_wait_asynccnt_4() {
  asm volatile("s_wait_asynccnt 4" ::: "memory");
}
__device__ __forceinline__ void s_wait_asynccnt_0() {
  asm volatile("s_wait_asynccnt 0" ::: "memory");
}

// ---- Kernel 1: per-chunk top-4 per (b, type, g) ---------------------------
// Block: 128 threads (4 wave32). Thread t owns type = t>>6 (0=pred cost,
// 1=anchor cost) and gt index g = t&63. Each wave's lanes read the same LDS
// address per step -> broadcast, no bank conflicts.
extern "C" __global__ void __launch_bounds__(128)
matcher_partial_topk(const float* __restrict__ pred,
                     const float* __restrict__ anc,
                     const float* __restrict__ gt,
                     float* __restrict__ wsv,
                     int*   __restrict__ wsi)
{
  __shared__ float4 sbuf[2][2][SQ];   // [double-buffer][pred/anc][query]

  const int t     = threadIdx.x;
  const int b     = blockIdx.y;
  const int chunk = blockIdx.x;
  const int type  = t >> 6;
  const int g     = t & (GN - 1);

  // GT box -> cxcywh held in registers
  const float* gp = gt + (size_t)(b * GN + g) * 4;
  const float gx0 = gp[0], gy0 = gp[1], gx1 = gp[2], gy1 = gp[3];
  const float gcx = (gx0 + gx1) * 0.5f;
  const float gcy = (gy0 + gy1) * 0.5f;
  const float gw  = gx1 - gx0;
  const float gh  = gy1 - gy0;

  const size_t qbase0 = ((size_t)b * QN + (size_t)chunk * CQ);
  const float* pbase  = pred + qbase0 * 4;
  const float* abase  = anc  + qbase0 * 4;

  // Issue one stage of async copies: 4 instructions per wave
  auto issue_stage = [&](int st) {
    const int bi = st & 1;
#pragma unroll
    for (int r = 0; r < 2; ++r) {
      const int ql = r * 128 + t;          // local query in stage
      const int qs = st * SQ + ql;         // query within chunk
      async_copy_b128(lds_off32(&sbuf[bi][0][ql]), pbase + (size_t)qs * 4);
      async_copy_b128(lds_off32(&sbuf[bi][1][ql]), abase + (size_t)qs * 4);
    }
  };

  issue_stage(0);

  float v0 = __builtin_inff(), v1 = v0, v2 = v0, v3 = v0;
  int   i0 = 0, i1 = 0, i2 = 0, i3 = 0;

  for (int s = 0; s < NST; ++s) {
    if (s + 1 < NST) {            // double-buffer: overlap next stage's DMA
      issue_stage(s + 1);
      s_wait_asynccnt_4();        // oldest 4 (stage s) complete, in order
    } else {
      s_wait_asynccnt_0();
    }
    __syncthreads();              // all waves' stage-s data visible in LDS

    const float4* qv  = &sbuf[s & 1][type][0];
    const int    qoff = chunk * CQ + s * SQ;

    for (int j = 0; j < SQ; ++j) {
      const float4 p = qv[j];     // uniform address -> LDS broadcast
      const float c = fabsf((p.x + p.z) * 0.5f - gcx)
                    + fabsf((p.y + p.w) * 0.5f - gcy)
                    + fabsf((p.z - p.x) - gw)
                    + fabsf((p.w - p.y) - gh);
      if (c < v3) {               // rare: full sorted insert (stable, strict <)
        const int q = qoff + j;
        const bool b0 = c < v0, b1 = c < v1, b2 = c < v2;
        v3 = b2 ? v2 : c;  i3 = b2 ? i2 : q;
        v2 = b2 ? (b1 ? v1 : c) : v2;  i2 = b2 ? (b1 ? i1 : q) : i2;
        v1 = b1 ? (b0 ? v0 : c) : v1;  i1 = b1 ? (b0 ? i0 : q) : i1;
        v0 = b0 ? c : v0;              i0 = b0 ? q : i0;
      }
    }
    __syncthreads();              // buffer (s&1) free for reuse next+1 stage
  }

  const int task = (b * 2 + type) * GN + g;          // == global thread task id
  const int slot = (task * CHUNKS + chunk) * KK;
  wsv[slot + 0] = v0; wsv[slot + 1] = v1; wsv[slot + 2] = v2; wsv[slot + 3] = v3;
  wsi[slot + 0] = i0; wsi[slot + 1] = i1; wsi[slot + 2] = i2; wsi[slot + 3] = i3;
}

// ---- Kernel 2: merge chunk candidates, emit idx_i / idx_j -----------------
extern "C" __global__ void __launch_bounds__(256)
matcher_merge(const float* __restrict__ wsv,
              const int*   __restrict__ wsi,
              int*         __restrict__ out)
{
  const int t = blockIdx.x * 256 + threadIdx.x;   // 0..NTASK-1
  if (t >= NTASK) return;
  const int g    = t & (GN - 1);
  const int type = (t >> 6) & 1;
  const int b    = t >> 7;

  float v0 = __builtin_inff(), v1 = v0, v2 = v0, v3 = v0;
  int   i0 = 0, i1 = 0, i2 = 0, i3 = 0;

  const int base = t * CHUNKS * KK;
  for (int c = 0; c < CHUNKS * KK; ++c) {         // chunk-major: stable ties
    const float cv = wsv[base + c];
    const int   ci = wsi[base + c];
    if (cv < v3) {
      const bool b0 = cv < v0, b1 = cv < v1, b2 = cv < v2;
      v3 = b2 ? v2 : cv;  i3 = b2 ? i2 : ci;
      v2 = b2 ? (b1 ? v1 : cv) : v2;  i2 = b2 ? (b1 ? i1 : ci) : i2;
      v1 = b1 ? (b0 ? v0 : cv) : v1;  i1 = b1 ? (b0 ? i0 : ci) : i1;
      v0 = b0 ? cv : v0;              i0 = b0 ? ci : i0;
    }
  }

  // idx_i layout: [bs, k, 2, G] flattened -> offset ((m*2 + type)*G + g)
  const int obase = b * (KK * 2 * GN);            // b * 512
  out[obase + (0 * 2 + type) * GN + g] = i0;
  out[obase + (1 * 2 + type) * GN + g] = i1;
  out[obase + (2 * 2 + type) * GN + g] = i2;
  out[obase + (3 * 2 + type) * GN + g] = i3;

  // idx_j: flat position p -> p % 64 (since 512 % 64 == 0)
  const int jbase = BSZ * KK * 2 * GN;            // 8192
#pragma unroll
  for (int e = 0; e < 4; ++e) {
    const int p = t * 4 + e;
    out[jbase + p] = p & (GN - 1);
  }
}

// ---- Host launch ----------------------------------------------------------
extern "C" void kernel_launch(void* const* d_in, const int* in_sizes, int n_in,
                              void* d_out, int out_size, void* d_ws, size_t ws_size,
                              hipStream_t stream) {
  const float* pred = (const float*)d_in[0];   // [16,32768,4] f32
  const float* anc  = (const float*)d_in[1];   // [16,32768,4] f32
  const float* gt   = (const float*)d_in[2];   // [1024,4] f32
  // d_in[3] (gt_labels) is unused by the reference outputs.

  float* wsv = (float*)d_ws;
  int*   wsi = (int*)((char*)d_ws + (size_t)NTASK * CHUNKS * KK * sizeof(float));

  dim3 grid1(CHUNKS, BSZ);
  matcher_partial_topk<<<grid1, 128, 0, stream>>>(pred, anc, gt, wsv, wsi);
  matcher_merge<<<NTASK / 256, 256, 0, stream>>>(wsv, wsi, (int*)d_out);
}